// TuningMode_1451698946238
// MI455X (gfx1250) — compile-verified
//
#include <hip/hip_runtime.h>
#include <hip/hip_bf16.h>
#include <math.h>

typedef __attribute__((ext_vector_type(2))) float v2f;
typedef __attribute__((ext_vector_type(8))) float v8f;

#define B_DIM 64
#define M_DIM 256
#define K_DIM 4096
#define EPS_C 0.0009f /* 0.03^2 */

// ---------------------------------------------------------------------------
// Kernel 1: row sums-of-squares -> norms.  Blocks 0..63 -> inputs rows,
// blocks 64..319 -> modes rows.  norms[] = [f_norm(64) | mv_norm(256)].
// ---------------------------------------------------------------------------
__global__ void norms_kernel(const float* __restrict__ inputs,
                             const float* __restrict__ modes,
                             float* __restrict__ norms) {
    __shared__ float red[256];
    const int row = blockIdx.x;
    const float* src = (row < B_DIM)
                           ? (inputs + (size_t)row * K_DIM)
                           : (modes + (size_t)(row - B_DIM) * K_DIM);
    float s = 0.0f;
    #pragma unroll
    for (int i = 0; i < K_DIM / 256; ++i) {
        float v = src[threadIdx.x + i * 256];   // coalesced
        s += v * v;
    }
    red[threadIdx.x] = s;
    __syncthreads();
    #pragma unroll
    for (int off = 128; off > 0; off >>= 1) {
        if ((int)threadIdx.x < off) red[threadIdx.x] += red[threadIdx.x + off];
        __syncthreads();
    }
    if (threadIdx.x == 0) norms[row] = sqrtf(red[0]);
}

// ---------------------------------------------------------------------------
// Kernel 2: dot = inputs @ modes^T via V_WMMA_F32_16X16X4_F32, fused metric.
//   A-tile (16x4 f32): lane l holds A[l&15][ (l>>4)*2 + {0,1} ] in 2 VGPRs.
//   B-tile (4x16 f32): lane l holds B[(l>>4)*2 + {0,1}][l&15]  -> since
//   B = modes^T, that is modes[n0 + (l&15)][k + (l>>4)*2 + {0,1}].
//   Both operands: one 8-byte load per lane per K-step of 4.
//   C/D (16x16 f32, 8 VGPRs): VGPR v = row v + 8*(l>>4), col = l&15.
// 128 threads = 4 waves; wave w handles n-tile (blockIdx>>2)*4 + w,
// m-tile = blockIdx&3.  EXEC all-ones inside each wave as required.
// ---------------------------------------------------------------------------
__global__ void wmma_metric_kernel(const float* __restrict__ A,   // [64,4096]
                                   const float* __restrict__ Bm,  // [256,4096]
                                   const float* __restrict__ norms,
                                   float* __restrict__ metric) {  // [64,256]
    const int lane  = threadIdx.x & 31;
    const int wave  = threadIdx.x >> 5;          // 0..3
    const int mTile = blockIdx.x & 3;            // 0..3   (batch tiles)
    const int nTile = ((blockIdx.x >> 2) << 2) | wave; // 0..15 (mode tiles)
    const int r16   = lane & 15;
    const int half  = lane >> 4;                 // 0 or 1

    const float* aRow = A  + (size_t)(mTile * 16 + r16) * K_DIM + half * 2;
    const float* bRow = Bm + (size_t)(nTile * 16 + r16) * K_DIM + half * 2;

    v8f acc = {};
    for (int k = 0; k < K_DIM; k += 32) {
        #pragma unroll
        for (int kk = 0; kk < 32; kk += 4) {
            v2f a = *(const v2f*)(aRow + k + kk);
            v2f b = *(const v2f*)(bRow + k + kk);
            // D = A x B + C ; emits v_wmma_f32_16x16x4_f32
            acc = __builtin_amdgcn_wmma_f32_16x16x4_f32(
                /*neg_a=*/false, a, /*neg_b=*/false, b,
                /*c_mod=*/(short)0, acc, /*reuse_a=*/false, /*reuse_b=*/false);
        }
    }

    // Fused epilogue: luminance / distance / metric per D element.
    const int n  = nTile * 16 + r16;       // mode index (column)
    const float mv = norms[B_DIM + n];
    #pragma unroll
    for (int v = 0; v < 8; ++v) {
        const int br   = mTile * 16 + half * 8 + v;   // batch index (row)
        const float f  = norms[br];
        const float dotv = acc[v];
        const float cosv = dotv / (f * mv);
        const float dnorm = sqrtf(fmaxf(2.0f - 2.0f * cosv, 0.0f));
        const float dist  = (2.0f - dnorm) * 0.5f;
        const float lum   = (2.0f * f * mv + EPS_C) / (f * f + mv * mv + EPS_C);
        const float met   = (1.0f - dist * sqrtf(lum)) * 2.0f;
        metric[(size_t)br * M_DIM + n] = met;
    }
}

// ---------------------------------------------------------------------------
// Kernel 3: out[b] = min_m metric[b][m]
// ---------------------------------------------------------------------------
__global__ void min_kernel(const float* __restrict__ metric,
                           float* __restrict__ out) {
    __shared__ float red[256];
    const int b = blockIdx.x;
    red[threadIdx.x] = metric[(size_t)b * M_DIM + threadIdx.x];
    __syncthreads();
    #pragma unroll
    for (int off = 128; off > 0; off >>= 1) {
        if ((int)threadIdx.x < off)
            red[threadIdx.x] = fminf(red[threadIdx.x], red[threadIdx.x + off]);
        __syncthreads();
    }
    if (threadIdx.x == 0) out[b] = red[0];
}

extern "C" void kernel_launch(void* const* d_in, const int* in_sizes, int n_in,
                              void* d_out, int out_size, void* d_ws, size_t ws_size,
                              hipStream_t stream) {
    const float* inputs = (const float*)d_in[0];   // [64,64,64]
    const float* modes  = (const float*)d_in[1];   // [256,64,64]
    float* out    = (float*)d_out;                 // [64]
    float* norms  = (float*)d_ws;                  // 320 floats
    float* metric = norms + (B_DIM + M_DIM);       // 64*256 floats

    norms_kernel<<<B_DIM + M_DIM, 256, 0, stream>>>(inputs, modes, norms);
    wmma_metric_kernel<<<16, 128, 0, stream>>>(inputs, modes, norms, metric);
    min_kernel<<<B_DIM, 256, 0, stream>>>(metric, out);
}